// SplatterModel_57947698758219
// MI455X (gfx1250) — compile-verified
//
#include <hip/hip_runtime.h>

#define NSPL 2048
#define H 128
#define W 128
#define EPS 1e-4f
#define CREC 8        // floats per coefficient record: a,b,c,d,e,f,pad,pad (32B)
#define CHUNK 128     // splats staged per LDS chunk

typedef float v2f __attribute__((ext_vector_type(2)));
typedef float v8f __attribute__((ext_vector_type(8)));

__device__ __forceinline__ float sigmoidf_(float x) { return 1.0f / (1.0f + __expf(-x)); }

// ---------------------------------------------------------------------------
// Kernel 1: per-splat polynomial coefficients + exact discrete max.
// One block (128 threads) per splat; thread t owns pixel row gy = t/W.
// Row max of the 1-D quadratic z(gx) is exact from {0, 127, floor(x*), ceil(x*)}.
// ---------------------------------------------------------------------------
__global__ __launch_bounds__(128) void splat_prep(
    const float* __restrict__ rho, const float* __restrict__ sigma,
    const float* __restrict__ coords, const float* __restrict__ alpha,
    const float* __restrict__ colors, const float* __restrict__ T,
    float* __restrict__ coefT, float* __restrict__ colT)
{
  const int n = blockIdx.x;
  const int t = threadIdx.x;

  // activations (redundant per thread; broadcast loads)
  const float r    = sigmoidf_(rho[n]) + EPS;
  const float aact = sigmoidf_(alpha[n]);
  const float s0 = sigmoidf_(-sigma[n * 3 + 0]);
  const float s1 = sigmoidf_(-sigma[n * 3 + 1]);
  const float s2 = sigmoidf_(-sigma[n * 3 + 2]);
  const float t0 = tanhf(coords[n * 3 + 0]) + 0.5f;
  const float t1 = tanhf(coords[n * 3 + 1]) + 0.5f;
  const float t2 = tanhf(coords[n * 3 + 2]) + 0.5f;

  // homogeneous transform (general 4x4; identity in the reference setup)
  const float sx = T[0] * s0 + T[1] * s1 + T[2]  * s2 + T[3];
  const float sy = T[4] * s0 + T[5] * s1 + T[6]  * s2 + T[7];
  const float sz = T[8] * s0 + T[9] * s1 + T[10] * s2 + T[11];
  const float cx = T[0] * t0 + T[1] * t1 + T[2]  * t2 + T[3];
  const float cy = T[4] * t0 + T[5] * t1 + T[6]  * t2 + T[7];
  const float cz = T[8] * t0 + T[9] * t1 + T[10] * t2 + T[11];
  const float w3 = 1.0f + cz;

  // symmetric covariance and M = -0.5 * inv(cov); logdet/log2pi cancel in
  // exp(k - k.max), and the downcast clip is a no-op for values in (0,1].
  const float c00 = sx * sx + EPS, c11 = sy * sy + EPS, c22 = sz * sz + EPS;
  const float c01 = sy * sx * r,  c02 = sz * sx * r,  c12 = sz * sy * r;
  const float d0  = c11 * c22 - c12 * c12;
  const float d1  = c01 * c22 - c12 * c02;
  const float d2  = c01 * c12 - c11 * c02;
  const float det = c00 * d0 - c01 * d1 + c02 * d2;
  const float idet = -0.5f / det;
  const float m00 = d0 * idet;
  const float m01 = (c02 * c12 - c01 * c22) * idet;
  const float m02 = d2 * idet;
  const float m11 = (c00 * c22 - c02 * c02) * idet;
  const float m12 = (c01 * c02 - c00 * c12) * idet;
  const float m22 = (c00 * c11 - c01 * c01) * idet;

  // z(gx,gy) = A gx^2 + B gx gy + C gy^2 + D gx + E gy + F
  const float A_ = m00;
  const float B_ = 2.0f * m01;
  const float C_ = m11;
  const float D_ = 2.0f * (m00 * cx + m01 * cy + m02 * w3);
  const float E_ = 2.0f * (m01 * cx + m11 * cy + m12 * w3);
  const float F_ = m00 * cx * cx + 2.0f * m01 * cx * cy + 2.0f * m02 * cx * w3 +
                   m11 * cy * cy + 2.0f * m12 * cy * w3 + m22 * w3 * w3;

  // exact row max: z_row(gx) = A gx^2 + b gx + c with gx = x/W, x in [0,127]
  const float invW = 1.0f / (float)W;
  const float gy = (float)t * invW;
  const float b_row = fmaf(B_, gy, D_);
  const float c_row = fmaf(fmaf(C_, gy, E_), gy, F_);
  // continuous vertex in integer-x units; IEEE max/min clamp handles NaN/Inf
  const float xf = (-b_row / (2.0f * A_)) * (float)W;
  const float xc = fminf(fmaxf(xf, 0.0f), 127.0f);
  const float x0 = floorf(xc);
  const float x1 = fminf(x0 + 1.0f, 127.0f);
  float zmax;
  {
    const float g0 = 0.0f, g1 = 127.0f * invW, g2 = x0 * invW, g3 = x1 * invW;
    const float za = fmaf(fmaf(A_, g0, b_row), g0, c_row);
    const float zb = fmaf(fmaf(A_, g1, b_row), g1, c_row);
    const float zc = fmaf(fmaf(A_, g2, b_row), g2, c_row);
    const float zd = fmaf(fmaf(A_, g3, b_row), g3, c_row);
    zmax = fmaxf(fmaxf(za, zb), fmaxf(zc, zd));
  }

  __shared__ float red[128];
  red[t] = zmax;
  __syncthreads();
  for (int s = 64; s > 0; s >>= 1) {
    if (t < s) red[t] = fmaxf(red[t], red[t + s]);
    __syncthreads();
  }

  if (t == 0) {
    float* rec = coefT + n * CREC;
    rec[0] = A_; rec[1] = B_; rec[2] = C_;
    rec[3] = D_; rec[4] = E_; rec[5] = F_ - red[0];   // fold max into constant
    rec[6] = 0.0f; rec[7] = 0.0f;
    colT[n * 4 + 0] = tanhf(colors[n * 4 + 0] * aact);
    colT[n * 4 + 1] = tanhf(colors[n * 4 + 1] * aact);
    colT[n * 4 + 2] = tanhf(colors[n * 4 + 2] * aact);
    colT[n * 4 + 3] = tanhf(colors[n * 4 + 3] * aact);
  }
}

// ---------------------------------------------------------------------------
// Kernel 2: render. out[pix,c] = sum_n exp(z'(n,pix)) * col[n,c]
// 8 waves/block; each wave owns 16 pixels (WMMA M rows). K-loop over splats,
// 4 per V_WMMA_F32_16X16X4_F32. Coefficients staged in LDS; the B operand is
// prebuilt in WMMA lane layout during staging and loaded FIRST in the body so
// the z/exp chain covers its LDS latency before the wmma consumes it.
// ---------------------------------------------------------------------------
__global__ __launch_bounds__(256) void splat_render(
    const float* __restrict__ coefT, const float* __restrict__ colT,
    float* __restrict__ out)
{
  __shared__ float clds[CHUNK * CREC];         // 4 KB coefficients
  __shared__ float blds[(CHUNK / 4) * 32 * 2]; // 8 KB B operands, lane layout

  const int t    = threadIdx.x;
  const int lane = t & 31;
  const int wv   = t >> 5;                     // 0..7
  const int m    = lane & 15;                  // pixel row within 16-tile / N col
  const int kh   = lane >> 4;                  // K half (0: K=0,1  1: K=2,3)
  const int pixel_base = blockIdx.x * 128 + wv * 16;
  const int pixel = pixel_base + m;

  const float invW = 1.0f / (float)W;
  const float gx = (float)(pixel & (W - 1)) * invW;
  const float gy = (float)(pixel >> 7) * invW;

  v8f acc = {};

  for (int ch = 0; ch < NSPL / CHUNK; ++ch) {
    __syncthreads();
    // stage coefficients
    {
      const float* src = coefT + ch * (CHUNK * CREC);
      for (int i = t; i < CHUNK * CREC; i += 256) clds[i] = src[i];
    }
    // stage B operands in WMMA 4x16 lane layout (zero-padded cols 4..15)
    for (int idx = t; idx < (CHUNK / 4) * 32; idx += 256) {
      const int g   = idx >> 5;                // K-group of 4 splats
      const int ln  = idx & 31;
      const int khh = ln >> 4;
      const int mm  = ln & 15;
      const int sp  = ch * CHUNK + g * 4 + 2 * khh;
      const float msk = (mm < 4) ? 1.0f : 0.0f;
      const int cc = mm & 3;
      blds[idx * 2 + 0] = colT[sp * 4 + cc] * msk;
      blds[idx * 2 + 1] = colT[(sp + 1) * 4 + cc] * msk;
    }
    __syncthreads();
    if (ch + 1 < NSPL / CHUNK) {               // global_prefetch_b8 of next chunk
      __builtin_prefetch(coefT + (ch + 1) * (CHUNK * CREC) + t, 0, 0);
      __builtin_prefetch(colT + (ch + 1) * (CHUNK * 4) + t, 0, 0);
    }

    #pragma unroll 8
    for (int k = 0; k < CHUNK; k += 4) {
      // issue the B-operand load first: its latency is hidden by the z chain
      const v2f Bop = *(const v2f*)&blds[((k >> 2) * 32 + lane) * 2];

      const float* r0 = &clds[(k + 2 * kh) * CREC];
      const float* r1 = r0 + CREC;
      // z for this lane's two (pixel, splat) pairs: 5 FMAs each
      const float u0 = fmaf(r0[0], gx, fmaf(r0[1], gy, r0[3]));
      const float v0 = fmaf(r0[2], gy, r0[4]);
      const float z0 = fmaf(u0, gx, fmaf(v0, gy, r0[5]));
      const float u1 = fmaf(r1[0], gx, fmaf(r1[1], gy, r1[3]));
      const float v1 = fmaf(r1[2], gy, r1[4]);
      const float z1 = fmaf(u1, gx, fmaf(v1, gy, r1[5]));

      v2f Aop;                                 // 16x4 A: exp kernel values
      Aop.x = __expf(z0);
      Aop.y = __expf(z1);

      acc = __builtin_amdgcn_wmma_f32_16x16x4_f32(
          false, Aop, false, Bop, (short)0, acc, false, false);
    }
  }

  // C/D layout: VGPR j -> pixel row (j | kh*8); lane%16 -> channel column
  if (m < 4) {
    const int prow = pixel_base + kh * 8;
    #pragma unroll
    for (int j = 0; j < 8; ++j)
      out[(prow + j) * 4 + m] = acc[j];
  }
}

extern "C" void kernel_launch(void* const* d_in, const int* in_sizes, int n_in,
                              void* d_out, int out_size, void* d_ws, size_t ws_size,
                              hipStream_t stream) {
  const float* rho       = (const float*)d_in[0];
  const float* sigma     = (const float*)d_in[1];
  const float* coords    = (const float*)d_in[2];
  const float* alpha     = (const float*)d_in[3];
  const float* colors    = (const float*)d_in[4];
  const float* transform = (const float*)d_in[5];
  float* out   = (float*)d_out;
  float* coefT = (float*)d_ws;                        // NSPL*8 floats = 64 KB
  float* colT  = (float*)d_ws + NSPL * CREC;          // NSPL*4 floats = 32 KB

  splat_prep<<<NSPL, 128, 0, stream>>>(rho, sigma, coords, alpha, colors,
                                       transform, coefT, colT);
  splat_render<<<(H * W) / 128, 256, 0, stream>>>(coefT, colT, out);
}